// MSDeformableAttention3D_7052336300061
// MI455X (gfx1250) — compile-verified
//
#include <hip/hip_runtime.h>
#include <hip/hip_bf16.h>
#include <math.h>

#define NUM_HEADS  8
#define NUM_LEVELS 4
#define NUM_POINTS 4
#define EMBED      256
#define QT         16        // queries per block
#define NOFF       256       // sampling-offset GEMM columns
#define NTOT       384       // + 128 attention-logit columns (8 heads * 16 kv)
#define SC_PITCH   392       // padded LDS pitch for C staging

typedef __attribute__((ext_vector_type(2)))  float v2f;
typedef __attribute__((ext_vector_type(8)))  float v8f;

__global__ __launch_bounds__(256)
void msda3d_fused_kernel(const float* __restrict__ query,
                         const float* __restrict__ key,
                         const float* __restrict__ value,
                         const float* __restrict__ refp,
                         const int*   __restrict__ sshape,
                         const float* __restrict__ W_so,
                         const float* __restrict__ b_so,
                         float* __restrict__ out)
{
    __shared__ float sC[QT][SC_PITCH];   // GEMM results: 16 queries x (256 off + 128 logits)
    __shared__ float sRowW[256][17];     // per-thread value-row weights (padded vs bank conflicts)

    const int tid    = threadIdx.x;
    const int wave   = tid >> 5;
    const int lane   = tid & 31;
    const int laneLo = lane & 15;
    const int laneHi = lane >> 4;
    const int q0     = blockIdx.x * QT;

    const float* qrow = query + (size_t)(q0 + laneLo) * EMBED;

    // Column tiling (branch-free B fragments):
    //   wave w owns N-tiles {w, w+8} from W_so and key-block tile 16+w (head w).
    const int c0 = wave * 16 + laneLo;          // W_so cols   0..127
    const int c1 = 128 + wave * 16 + laneLo;    // W_so cols 128..255
    const float* krow = key + (size_t)laneLo * EMBED;   // kv = laneLo for key tile

    v8f acc0 = {}; v8f acc1 = {}; v8f acc2 = {};

    // ---- Main GEMM: 2 pure-W_so tiles per wave, f32 WMMA, K tiled by 4 ----
    // A 16x4 f32 frag: lane row = laneLo, VGPR{0,1} hold K = laneHi*2 + {0,1}.
    // B 4x16 f32 frag: lane col = laneLo, VGPR{0,1} hold K = laneHi*2 + {0,1}.
    #pragma unroll 4
    for (int k0 = 0; k0 < EMBED; k0 += 4) {
        const int ka = k0 + laneHi * 2;
        const v2f A = *(const v2f*)(qrow + ka);                 // global_load_b64
        const float* wr0 = W_so + (size_t)ka * NOFF;
        const float* wr1 = W_so + (size_t)(ka + 1) * NOFF;
        v2f B0; B0[0] = wr0[c0]; B0[1] = wr1[c0];
        v2f B1; B1[0] = wr0[c1]; B1[1] = wr1[c1];
        acc0 = __builtin_amdgcn_wmma_f32_16x16x4_f32(false, A, false, B0, (short)0, acc0, false, false);
        acc1 = __builtin_amdgcn_wmma_f32_16x16x4_f32(false, A, false, B1, (short)0, acc1, false, false);
    }

    // ---- Attention-logit tile: head == wave, B nonzero only for K in [32w, 32w+32) ----
    {
        const int kbeg = wave * 32;
        #pragma unroll
        for (int kk = 0; kk < 32; kk += 4) {
            const int ka = kbeg + kk + laneHi * 2;
            const v2f A  = *(const v2f*)(qrow + ka);            // L1 hit (reload)
            const v2f B2 = *(const v2f*)(krow + ka);            // global_load_b64
            acc2 = __builtin_amdgcn_wmma_f32_16x16x4_f32(false, A, false, B2, (short)0, acc2, false, false);
        }
    }

    // Stage C tiles to LDS: C layout lane = N + 16*(M>=8), vgpr r = M%8.
    #pragma unroll
    for (int r = 0; r < 8; ++r) {
        const int m = r + laneHi * 8;
        sC[m][c0]       = acc0[r];
        sC[m][c1]       = acc1[r];
        sC[m][NOFF + wave * 16 + laneLo] = acc2[r];
    }
    __syncthreads();

    // ---------------- epilogue: 2 threads per (query, head) ----------------
    const int q    = tid >> 4;        // 0..15
    const int rem  = tid & 15;
    const int h    = rem >> 1;        // 0..7
    const int half = rem & 1;         // low/high 16 channels of d=32
    const int qg   = q0 + q;

    int Hs[4], Ws[4], S[4];
    {
        int s = 0;
        #pragma unroll
        for (int L = 0; L < 4; ++L) {
            Hs[L] = sshape[2 * L];
            Ws[L] = sshape[2 * L + 1];
            S[L]  = s;
            s += Hs[L] * Ws[L];
        }
    }

    // softmax over the 16 kv logits of this head
    float lg[16];
    float mx = -3.0e38f;
    #pragma unroll
    for (int i = 0; i < 16; ++i) { lg[i] = sC[q][NOFF + h * 16 + i]; mx = fmaxf(mx, lg[i]); }
    float se = 0.0f;
    #pragma unroll
    for (int i = 0; i < 16; ++i) { lg[i] = __expf(lg[i] - mx); se += lg[i]; }
    const float inv = 1.0f / se;

    #pragma unroll
    for (int r = 0; r < 16; ++r) sRowW[tid][r] = 0.0f;

    // Faithful to the reference's (p,l)->(l,p) reshape: the level-L sample uses
    // ref point / normalization of level P and offset index ((h*4+L)*4+P)*2.
    #pragma unroll
    for (int L = 0; L < 4; ++L) {
        const float Wl = (float)Ws[L], Hl = (float)Hs[L];
        const int   Wi = Ws[L], Hi = Hs[L], Sl = S[L];
        #pragma unroll
        for (int P = 0; P < 4; ++P) {
            const float a  = lg[L * 4 + P] * inv;
            const int   n  = ((h * 4 + L) * 4 + P) * 2;
            const float ox = sC[q][n]     + b_so[n];
            const float oy = sC[q][n + 1] + b_so[n + 1];
            const float rx = refp[((size_t)qg * 4 + P) * 2 + 0];
            const float ry = refp[((size_t)qg * 4 + P) * 2 + 1];
            const float locx = rx + ox / (float)Ws[P];
            const float locy = ry + oy / (float)Hs[P];
            // grid = 2*loc - 1; x = ((gx+1)*W - 1)*0.5 = loc*W - 0.5
            const float x  = locx * Wl - 0.5f;
            const float y  = locy * Hl - 0.5f;
            const float x0f = floorf(x), y0f = floorf(y);
            const int   x0  = (int)x0f,  y0  = (int)y0f;
            const float fx  = x - x0f,   fy  = y - y0f;
            const float w00 = (1.0f - fx) * (1.0f - fy) * a;
            const float w10 = fx * (1.0f - fy) * a;
            const float w01 = (1.0f - fx) * fy * a;
            const float w11 = fx * fy * a;
            const int x1 = x0 + 1, y1 = y0 + 1;
            if (x0 >= 0 && x0 < Wi && y0 >= 0 && y0 < Hi) sRowW[tid][Sl + y0 * Wi + x0] += w00;
            if (x1 >= 0 && x1 < Wi && y0 >= 0 && y0 < Hi) sRowW[tid][Sl + y0 * Wi + x1] += w10;
            if (x0 >= 0 && x0 < Wi && y1 >= 0 && y1 < Hi) sRowW[tid][Sl + y1 * Wi + x0] += w01;
            if (x1 >= 0 && x1 < Wi && y1 >= 0 && y1 < Hi) sRowW[tid][Sl + y1 * Wi + x1] += w11;
        }
    }

    // Dense 16-row x 16-channel accumulation from the (L1-resident) value tensor.
    const int cbase = h * 32 + half * 16;
    float accv[16];
    #pragma unroll
    for (int c = 0; c < 16; ++c) accv[c] = 0.0f;
    #pragma unroll
    for (int r = 0; r < 16; ++r) {
        const float wgt = sRowW[tid][r];
        const float* vr = value + (size_t)r * EMBED + cbase;
        #pragma unroll
        for (int c = 0; c < 16; ++c) accv[c] = fmaf(wgt, vr[c], accv[c]);
    }

    float* op = out + (size_t)qg * EMBED + cbase;
    #pragma unroll
    for (int c = 0; c < 16; ++c) op[c] = accv[c];
}

extern "C" void kernel_launch(void* const* d_in, const int* in_sizes, int n_in,
                              void* d_out, int out_size, void* d_ws, size_t ws_size,
                              hipStream_t stream) {
    (void)n_in; (void)out_size; (void)d_ws; (void)ws_size;
    const float* query = (const float*)d_in[0];
    const float* key   = (const float*)d_in[1];
    const float* value = (const float*)d_in[2];
    const float* refp  = (const float*)d_in[3];
    const int*   ss    = (const int*)  d_in[4];
    const float* W_so  = (const float*)d_in[5];
    const float* b_so  = (const float*)d_in[6];
    float*       out   = (float*)d_out;

    const int num_q  = in_sizes[0] / EMBED;          // 40000
    const int blocks = (num_q + QT - 1) / QT;        // 2500
    hipLaunchKernelGGL(msda3d_fused_kernel, dim3(blocks), dim3(256), 0, stream,
                       query, key, value, refp, ss, W_so, b_so, out);
}